// VQVAE_36807869727047
// MI455X (gfx1250) — compile-verified
//
#include <hip/hip_runtime.h>

typedef float v2f __attribute__((ext_vector_type(2)));
typedef float v8f __attribute__((ext_vector_type(8)));

#define NXR    8192          // x rows (8*32*32)
#define KCB    8192          // codebook entries
#define DIMD   256
#define INV_T  100.0f        // 1/ENTROPY_TEMPERATURE

// workspace layout (float units)
#define A2_OFF   0                   // a2[8192]
#define B2_OFF   8192                // b2[8192]
#define PACK_OFF 16384               // float4 {a2, m, log s, 1/s} per x-row
#define IMIN_OFF 49152               // int argmin per x-row
#define AVGP_OFF 57344               // avg_probs[8192]
#define SC_OFF   65536               // [0]=mse sum, [1]=sum p*logp

#define QOFF     2097152             // d_out: quantized (2097152) | 4 scalars | 8192 indices

// Pipelined 16x16x(full-256) fp32 WMMA macro-tile.
// K is re-mapped: lane-half 0 consumes k in [0,128), half 1 consumes [128,256),
// identically for A and B, so the contraction is the exact dot product while
// every lane streams 128 contiguous floats -> b128 loads, 2 k-steps per load.
__device__ __forceinline__ v8f gemm_tile_256(const float4* __restrict__ a4,
                                             const float4* __restrict__ b4) {
  v8f acc = {0.f,0.f,0.f,0.f,0.f,0.f,0.f,0.f};
  float4 aP[4], bP[4];
  #pragma unroll
  for (int i = 0; i < 4; ++i) { aP[i] = a4[i]; bP[i] = b4[i]; }
  #pragma unroll
  for (int jj = 0; jj < 32; ++jj) {
    float4 ac = aP[jj & 3], bc = bP[jj & 3];
    v2f al = {ac.x, ac.y}, ah = {ac.z, ac.w};
    v2f bl = {bc.x, bc.y}, bh = {bc.z, bc.w};
    acc = __builtin_amdgcn_wmma_f32_16x16x4_f32(false, al, false, bl,
                                                (short)0, acc, false, false);
    acc = __builtin_amdgcn_wmma_f32_16x16x4_f32(false, ah, false, bh,
                                                (short)0, acc, false, false);
    if (jj + 4 < 32) { aP[jj & 3] = a4[jj + 4]; bP[jj & 3] = b4[jj + 4]; }
  }
  return acc;
}

// ---------------------------------------------------------------- norms + init
__global__ __launch_bounds__(256) void vq_norms(const float* __restrict__ x,
                                                const float* __restrict__ cb,
                                                float* __restrict__ ws) {
  if (blockIdx.x == 0 && threadIdx.x == 0) { ws[SC_OFF] = 0.0f; ws[SC_OFF + 1] = 0.0f; }
  int w = threadIdx.x >> 5, lane = threadIdx.x & 31;
  int row = blockIdx.x * 8 + w;                      // 0..16383
  const float* src = (row < NXR) ? (x + (size_t)row * DIMD)
                                 : (cb + (size_t)(row - NXR) * DIMD);
  float4 p0 = *(const float4*)(src + lane * 8);
  float4 p1 = *(const float4*)(src + lane * 8 + 4);
  float s = p0.x*p0.x + p0.y*p0.y + p0.z*p0.z + p0.w*p0.w
          + p1.x*p1.x + p1.y*p1.y + p1.z*p1.z + p1.w*p1.w;
  #pragma unroll
  for (int msk = 1; msk < 32; msk <<= 1) s += __shfl_xor(s, msk, 32);
  if (lane == 0) {
    if (row < NXR) ws[A2_OFF + row] = s;
    else           ws[B2_OFF + row - NXR] = s;
  }
}

// ------------------------------------------- pass 1: fused GEMM + online stats
// WG = 32 x-rows (2 row-tiles) x all K. 8 waves: rt = w&1, parity = w>>1 (4-way K split).
__global__ __launch_bounds__(256) void vq_pass1(const float* __restrict__ x,
                                                const float* __restrict__ cb,
                                                float* __restrict__ ws) {
  __shared__ float sm[128], ssh[128], sdh[128];
  __shared__ int   sih[128];
  int tid = threadIdx.x, w = tid >> 5, lane = tid & 31;
  int half = lane >> 4, ln = lane & 15;
  int rt = w & 1, parity = w >> 1;
  int rowBase = blockIdx.x * 32 + rt * 16;

  // float4-unit base pointers with the k re-map (half picks 128-float range)
  const float4* a4 = (const float4*)x  + (size_t)(rowBase + ln) * 64 + 32 * half;
  const float4* b4base = (const float4*)cb + (size_t)ln * 64 + 32 * half;

  float a2r[8];
  #pragma unroll
  for (int r = 0; r < 8; ++r) a2r[r] = ws[A2_OFF + rowBase + r + 8 * half];

  float m[8], s[8], dmin[8]; int imin[8];
  #pragma unroll
  for (int r = 0; r < 8; ++r) {
    m[r] = -__builtin_inff(); s[r] = 0.0f; dmin[r] = __builtin_inff(); imin[r] = 0;
  }

  for (int ct = parity; ct < KCB / 16; ct += 4) {
    float b2c = ws[B2_OFF + ct * 16 + ln];
    const float4* b4 = b4base + (size_t)ct * 16 * 64;
    v8f acc = gemm_tile_256(a4, b4);
    int cbIdx = ct * 16 + ln;
    #pragma unroll
    for (int r = 0; r < 8; ++r) {
      float dist = a2r[r] + b2c - 2.0f * acc[r];
      bool lt = dist < dmin[r];
      dmin[r] = lt ? dist : dmin[r];
      imin[r] = lt ? cbIdx : imin[r];
      float l = -INV_T * dist;
      float d = l - m[r];
      float e = __expf(-fabsf(d));          // exp(-|d|), single transcendental
      bool gt = d > 0.0f;
      s[r] = gt ? (s[r] * e + 1.0f) : (s[r] + e);
      m[r] = gt ? l : m[r];
    }
  }

  // reduce across the 16 lanes of each half (columns of the tile)
  #pragma unroll
  for (int r = 0; r < 8; ++r) {
    #pragma unroll
    for (int msk = 1; msk < 16; msk <<= 1) {
      float mo = __shfl_xor(m[r], msk, 32);
      float so = __shfl_xor(s[r], msk, 32);
      float dn = __shfl_xor(dmin[r], msk, 32);
      int   io = __shfl_xor(imin[r], msk, 32);
      float M = fmaxf(m[r], mo);
      s[r] = s[r] * __expf(m[r] - M) + so * __expf(mo - M);
      m[r] = M;
      bool take = (dn < dmin[r]) || (dn == dmin[r] && io < imin[r]);
      dmin[r] = take ? dn : dmin[r];
      imin[r] = take ? io : imin[r];
    }
  }

  if (ln == 0) {  // lanes 0 and 16 hold the two row halves
    #pragma unroll
    for (int r = 0; r < 8; ++r) {
      int slot = (rt * 16 + r + 8 * half) * 4 + parity;
      sm[slot] = m[r]; ssh[slot] = s[r]; sdh[slot] = dmin[r]; sih[slot] = imin[r];
    }
  }
  __syncthreads();

  if (tid < 32) {  // merge the 4 parity partials per row, emit row pack + argmin
    int n = blockIdx.x * 32 + tid;
    float M = -__builtin_inff();
    #pragma unroll
    for (int p = 0; p < 4; ++p) M = fmaxf(M, sm[tid * 4 + p]);
    float S = 0.0f; float dbest = __builtin_inff(); int ibest = 0x7fffffff;
    #pragma unroll
    for (int p = 0; p < 4; ++p) {
      S += ssh[tid * 4 + p] * __expf(sm[tid * 4 + p] - M);
      float dp = sdh[tid * 4 + p]; int ip = sih[tid * 4 + p];
      bool take = (dp < dbest) || (dp == dbest && ip < ibest);
      dbest = take ? dp : dbest;
      ibest = take ? ip : ibest;
    }
    float4 pk = make_float4(ws[A2_OFF + n], M, __logf(S), 1.0f / S);
    *(float4*)(ws + PACK_OFF + (size_t)n * 4) = pk;
    ((int*)ws)[IMIN_OFF + n] = ibest;
  }
}

// ---------------- pass 2: transposed GEMM -> avg_probs columns + sample entropy
// WG = 32 codebook entries (2 tiles) x all N. kt = w&1, parity = w>>1.
__global__ __launch_bounds__(256) void vq_pass2(const float* __restrict__ x,
                                                const float* __restrict__ cb,
                                                float* __restrict__ ws) {
  __shared__ float colpart[32];
  int tid = threadIdx.x, w = tid >> 5, lane = tid & 31;
  int half = lane >> 4, ln = lane & 15;
  int kt = w & 1, parity = w >> 1;
  int kBase = blockIdx.x * 32 + kt * 16;
  if (tid < 32) colpart[tid] = 0.0f;
  __syncthreads();

  const float4* a4 = (const float4*)cb + (size_t)(kBase + ln) * 64 + 32 * half;
  const float4* b4base = (const float4*)x + (size_t)ln * 64 + 32 * half;

  float b2r[8];
  #pragma unroll
  for (int r = 0; r < 8; ++r) b2r[r] = ws[B2_OFF + kBase + r + 8 * half];

  float colacc[8];
  #pragma unroll
  for (int r = 0; r < 8; ++r) colacc[r] = 0.0f;
  float sH = 0.0f;

  for (int nt = parity; nt < NXR / 16; nt += 4) {
    int n = nt * 16 + ln;
    float4 pk = *(const float4*)(ws + PACK_OFF + (size_t)n * 4); // {a2, m, log s, 1/s}
    const float4* b4 = b4base + (size_t)nt * 16 * 64;
    v8f acc = gemm_tile_256(a4, b4);
    #pragma unroll
    for (int r = 0; r < 8; ++r) {
      float dist = pk.x + b2r[r] - 2.0f * acc[r];
      float l = -INV_T * dist;
      float p = __expf(l - pk.y) * pk.w;       // softmax prob
      colacc[r] += p;
      sH += p * (l - pk.y - pk.z);             // p * log p
    }
  }

  #pragma unroll
  for (int r = 0; r < 8; ++r) {
    #pragma unroll
    for (int msk = 1; msk < 16; msk <<= 1) colacc[r] += __shfl_xor(colacc[r], msk, 32);
  }
  if (ln == 0) {
    #pragma unroll
    for (int r = 0; r < 8; ++r)
      atomicAdd(&colpart[kt * 16 + r + 8 * half], colacc[r]);
  }
  #pragma unroll
  for (int msk = 1; msk < 32; msk <<= 1) sH += __shfl_xor(sH, msk, 32);
  if (lane == 0) atomicAdd(ws + SC_OFF + 1, sH);
  __syncthreads();
  if (tid < 32)
    ws[AVGP_OFF + blockIdx.x * 32 + tid] = colpart[tid] * (1.0f / (float)NXR);
}

// -------------------------- gather quantized, MSE sum, indices to output
__global__ __launch_bounds__(256) void vq_gather(const float* __restrict__ x,
                                                 const float* __restrict__ cb,
                                                 float* __restrict__ ws,
                                                 float* __restrict__ out) {
  int w = threadIdx.x >> 5, lane = threadIdx.x & 31;
  int n = blockIdx.x * 8 + w;
  int idx = ((const int*)ws)[IMIN_OFF + n];
  const float* q  = cb + (size_t)idx * DIMD;
  const float* xr = x  + (size_t)n * DIMD;
  float accum = 0.0f;
  #pragma unroll
  for (int j = 0; j < 2; ++j) {
    float4 qv = *(const float4*)(q  + lane * 8 + j * 4);
    float4 xv = *(const float4*)(xr + lane * 8 + j * 4);
    *(float4*)(out + (size_t)n * DIMD + lane * 8 + j * 4) = qv; // straight-through == quantized
    float dx = qv.x - xv.x, dy = qv.y - xv.y, dz = qv.z - xv.z, dw = qv.w - xv.w;
    accum += dx*dx + dy*dy + dz*dz + dw*dw;
  }
  #pragma unroll
  for (int msk = 1; msk < 32; msk <<= 1) accum += __shfl_xor(accum, msk, 32);
  if (lane == 0) {
    atomicAdd(ws + SC_OFF, accum);
    out[QOFF + 4 + n] = (float)idx;            // encoding_indices
  }
}

// ------------------------------------------------------------- finalize scalars
__global__ __launch_bounds__(256) void vq_finalize(const float* __restrict__ ws,
                                                   float* __restrict__ out) {
  __shared__ float red[256];
  float a = 0.0f;
  for (int k = threadIdx.x; k < KCB; k += 256) {
    float p = ws[AVGP_OFF + k];
    a -= p * __logf(p + 1e-5f);
  }
  red[threadIdx.x] = a;
  __syncthreads();
  for (int st = 128; st > 0; st >>= 1) {
    if (threadIdx.x < st) red[threadIdx.x] += red[threadIdx.x + st];
    __syncthreads();
  }
  if (threadIdx.x == 0) {
    float avgH    = red[0];
    float mse     = ws[SC_OFF] * (1.0f / (float)(NXR * DIMD));
    float sampleH = -ws[SC_OFF + 1] * (1.0f / (float)NXR);
    float ent     = (sampleH - avgH) * 0.1f;       // ENTROPY_LOSS_RATIO
    out[QOFF + 0] = 1.25f * mse + ent;             // loss
    out[QOFF + 1] = 0.25f * mse;                   // e_latent_loss
    out[QOFF + 2] = mse;                           // q_latent_loss
    out[QOFF + 3] = ent;                           // entropy_loss
  }
}

extern "C" void kernel_launch(void* const* d_in, const int* in_sizes, int n_in,
                              void* d_out, int out_size, void* d_ws, size_t ws_size,
                              hipStream_t stream) {
  (void)in_sizes; (void)n_in; (void)out_size; (void)ws_size;
  const float* x  = (const float*)d_in[0];
  const float* cb = (const float*)d_in[1];
  float* ws  = (float*)d_ws;
  float* out = (float*)d_out;

  vq_norms   <<<2048, 256, 0, stream>>>(x, cb, ws);
  vq_pass1   <<< 256, 256, 0, stream>>>(x, cb, ws);
  vq_pass2   <<< 256, 256, 0, stream>>>(x, cb, ws);
  vq_gather  <<<1024, 256, 0, stream>>>(x, cb, ws, out);
  vq_finalize<<<   1, 256, 0, stream>>>(ws, out);
}